// SparseDeepUnrolling_52948356825092
// MI455X (gfx1250) — compile-verified
//
#include <hip/hip_runtime.h>

typedef float v2f __attribute__((ext_vector_type(2)));
typedef float v8f __attribute__((ext_vector_type(8)));

#define B_ 4096
#define D_ 1024
#define P_ 2048
#define T_ 10
#define LAM_ 0.12f
#define STEP_ 0.1f

// ---------------------------------------------------------------------------
// Column-normalize the dictionary: Wn[:,p] = W[:,p] / ||W[:,p]||_2
// One thread per column (P columns); row-walk is coalesced across the wave.
// ---------------------------------------------------------------------------
__global__ __launch_bounds__(256) void colnorm_kernel(const float* __restrict__ W,
                                                      float* __restrict__ Wn) {
  const int p = blockIdx.x * 256 + threadIdx.x;
  float s = 0.0f;
  for (int d = 0; d < D_; ++d) {
    float v = W[(size_t)d * P_ + p];
    s = fmaf(v, v, s);
  }
  const float inv = 1.0f / sqrtf(s);
  for (int d = 0; d < D_; ++d) {
    Wn[(size_t)d * P_ + p] = W[(size_t)d * P_ + p] * inv;
  }
}

// z1 = relu(b)  (first LISTA step with z0 == 0), vectorized float4.
__global__ __launch_bounds__(256) void relu4_kernel(const float4* __restrict__ in,
                                                    float4* __restrict__ out, int n4) {
  const int i = blockIdx.x * 256 + threadIdx.x;
  if (i < n4) {
    float4 v = in[i];
    v.x = fmaxf(v.x, 0.0f);
    v.y = fmaxf(v.y, 0.0f);
    v.z = fmaxf(v.z, 0.0f);
    v.w = fmaxf(v.w, 0.0f);
    out[i] = v;
  }
}

// ---------------------------------------------------------------------------
// FP32 WMMA GEMM: C(MxN) = epilogue( opA(A) @ opB(B) )
//   TA: operand A element (m,k) = A[k*lda + m]   (else A[m*lda + k])
//   TB: operand B element (k,n) = B[n*ldb + k]   (else B[k*ldb + n])
//   EPI 0: C = acc
//   EPI 1: C = I - STEP*acc                (build A-matrix from Wn^T Wn)
//   EPI 2: C = STEP*acc - LAM*STEP         (bias b from x @ Wn)
//   EPI 3: C = relu(acc + bias)            (LISTA iteration, bias stride ldc)
// 128x128 block tile, 8 waves (4x2), each wave 32x64 = 2x4 tiles of 16x16.
// K staged through LDS in slabs of 32 (8 WMMA k-steps of 4), global side
// loaded as float4 (GLOBAL_LOAD_B128). All dims multiples of 128 -> no guards.
// LDS pad of +4 keeps float4 row stores 16B-aligned (stride 132 dwords).
// ---------------------------------------------------------------------------
template <bool TA, bool TB, int EPI>
__global__ __launch_bounds__(256)
void gemm_wmma_kernel(const float* __restrict__ A, const float* __restrict__ Bm,
                      const float* __restrict__ bias, float* __restrict__ C,
                      int M, int N, int K, int lda, int ldb, int ldc) {
  constexpr int BM = 128, BN = 128, BK = 32, PAD = 4;
  __shared__ float As[BK][BM + PAD];
  __shared__ float Bs[BK][BN + PAD];

  const int tid   = threadIdx.x;
  const int wave  = tid >> 5;
  const int lane  = tid & 31;
  const int waveM = wave >> 1;   // 0..3 -> 32-row strip
  const int waveN = wave & 1;    // 0..1 -> 64-col strip
  const int lhalf = lane >> 4;   // 0/1
  const int lmod  = lane & 15;
  const int bm = blockIdx.y * BM;
  const int bn = blockIdx.x * BN;

  v8f acc[2][4];
#pragma unroll
  for (int mt = 0; mt < 2; ++mt)
#pragma unroll
    for (int nt = 0; nt < 4; ++nt)
      acc[mt][nt] = (v8f){0.f, 0.f, 0.f, 0.f, 0.f, 0.f, 0.f, 0.f};

  for (int k0 = 0; k0 < K; k0 += BK) {
    // ---- stage A slab: As[k][m], 1024 float4s via 256 threads x 4 ----
#pragma unroll
    for (int i = 0; i < 4; ++i) {
      const int e = tid + i * 256;
      if constexpr (TA) {
        // global contiguous along m: float4 load + float4 LDS store
        const int k = e >> 5, m4 = (e & 31) * 4;
        const float4 v = *reinterpret_cast<const float4*>(
            &A[(size_t)(k0 + k) * lda + (bm + m4)]);
        *reinterpret_cast<float4*>(&As[k][m4]) = v;
      } else {
        // global contiguous along k: float4 load + 4 scattered LDS stores
        const int m = e >> 3, k4 = (e & 7) * 4;
        const float4 v = *reinterpret_cast<const float4*>(
            &A[(size_t)(bm + m) * lda + (k0 + k4)]);
        As[k4 + 0][m] = v.x;
        As[k4 + 1][m] = v.y;
        As[k4 + 2][m] = v.z;
        As[k4 + 3][m] = v.w;
      }
    }
    // ---- stage B slab: Bs[k][n] ----
#pragma unroll
    for (int i = 0; i < 4; ++i) {
      const int e = tid + i * 256;
      if constexpr (TB) {
        const int n = e >> 3, k4 = (e & 7) * 4;
        const float4 v = *reinterpret_cast<const float4*>(
            &Bm[(size_t)(bn + n) * ldb + (k0 + k4)]);
        Bs[k4 + 0][n] = v.x;
        Bs[k4 + 1][n] = v.y;
        Bs[k4 + 2][n] = v.z;
        Bs[k4 + 3][n] = v.w;
      } else {
        const int k = e >> 5, n4 = (e & 31) * 4;
        const float4 v = *reinterpret_cast<const float4*>(
            &Bm[(size_t)(k0 + k) * ldb + (bn + n4)]);
        *reinterpret_cast<float4*>(&Bs[k][n4]) = v;
      }
    }
    __syncthreads();

    // ---- 8 WMMA k-steps over the slab ----
#pragma unroll
    for (int kk = 0; kk < BK; kk += 4) {
      const int kr = kk + lhalf * 2;  // ISA frag: K = (lane>>4)*2 + v
      v2f af[2], bf[4];
#pragma unroll
      for (int mt = 0; mt < 2; ++mt) {
        const int m = waveM * 32 + mt * 16 + lmod;  // M = lane&15
        af[mt][0] = As[kr][m];
        af[mt][1] = As[kr + 1][m];
      }
#pragma unroll
      for (int nt = 0; nt < 4; ++nt) {
        const int n = waveN * 64 + nt * 16 + lmod;  // N = lane&15
        bf[nt][0] = Bs[kr][n];
        bf[nt][1] = Bs[kr + 1][n];
      }
#pragma unroll
      for (int mt = 0; mt < 2; ++mt)
#pragma unroll
        for (int nt = 0; nt < 4; ++nt)
          acc[mt][nt] = __builtin_amdgcn_wmma_f32_16x16x4_f32(
              false, af[mt], false, bf[nt], (short)0, acc[mt][nt], false, false);
    }
    __syncthreads();
  }

  // ---- epilogue + store (C/D layout: M = r + (lane>>4)*8, N = lane&15) ----
#pragma unroll
  for (int mt = 0; mt < 2; ++mt) {
#pragma unroll
    for (int nt = 0; nt < 4; ++nt) {
      const int gn = bn + waveN * 64 + nt * 16 + lmod;
#pragma unroll
      for (int r = 0; r < 8; ++r) {
        const int gm = bm + waveM * 32 + mt * 16 + lhalf * 8 + r;
        float v = acc[mt][nt][r];
        if (EPI == 1)      v = (gm == gn ? 1.0f : 0.0f) - STEP_ * v;
        else if (EPI == 2) v = STEP_ * v - LAM_ * STEP_;
        else if (EPI == 3) v = fmaxf(v + bias[(size_t)gm * ldc + gn], 0.0f);
        C[(size_t)gm * ldc + gn] = v;
      }
    }
  }
}

// ---------------------------------------------------------------------------
// Orchestration:
//   ws: Wn (8MB) | Amat (16MB) | bmat (32MB) | ztmp (32MB)
//   d_out: xhat (B*D) | z (B*P)   -> z ping-pongs between ztmp and d_out slice
// ---------------------------------------------------------------------------
extern "C" void kernel_launch(void* const* d_in, const int* in_sizes, int n_in,
                              void* d_out, int out_size, void* d_ws, size_t ws_size,
                              hipStream_t stream) {
  (void)in_sizes; (void)n_in; (void)out_size; (void)ws_size;
  const float* x = (const float*)d_in[0];  // (B, D, 1)
  const float* W = (const float*)d_in[1];  // (D, P)
  float* out = (float*)d_out;
  float* ws  = (float*)d_ws;

  float* Wn   = ws;                         // D*P
  float* Amat = Wn + (size_t)D_ * P_;       // P*P
  float* bmat = Amat + (size_t)P_ * P_;     // B*P
  float* ztmp = bmat + (size_t)B_ * P_;     // B*P
  float* zout = out + (size_t)B_ * D_;      // B*P slice of d_out
  float* xhat = out;                        // B*D slice of d_out

  // Wn = normalize(W, dim=0)
  colnorm_kernel<<<P_ / 256, 256, 0, stream>>>(W, Wn);

  // Amat = I - STEP * Wn^T @ Wn   (TA: A operand is Wn transposed)
  gemm_wmma_kernel<true, false, 1><<<dim3(P_ / 128, P_ / 128), 256, 0, stream>>>(
      Wn, Wn, nullptr, Amat, P_, P_, D_, P_, P_, P_);

  // bmat = STEP * (x @ Wn) - LAM*STEP
  gemm_wmma_kernel<false, false, 2><<<dim3(P_ / 128, B_ / 128), 256, 0, stream>>>(
      x, Wn, nullptr, bmat, B_, P_, D_, D_, P_, P_);

  // z1 = relu(bmat)
  relu4_kernel<<<(B_ * P_ / 4) / 256, 256, 0, stream>>>(
      (const float4*)bmat, (float4*)ztmp, B_ * P_ / 4);

  // 9 remaining iterations: z <- relu(z @ Amat + bmat), ping-pong ztmp<->zout.
  // Odd count (9) starting from ztmp => final z10 lands in zout (d_out slice).
  float* zsrc = ztmp;
  float* zdst = zout;
  for (int t = 1; t < T_; ++t) {
    gemm_wmma_kernel<false, false, 3><<<dim3(P_ / 128, B_ / 128), 256, 0, stream>>>(
        zsrc, Amat, bmat, zdst, B_, P_, P_, P_, P_, P_);
    float* tmp = zsrc; zsrc = zdst; zdst = tmp;
  }

  // xhat = z10 @ Wn^T  (TB: B operand is Wn transposed)
  gemm_wmma_kernel<false, true, 0><<<dim3(D_ / 128, B_ / 128), 256, 0, stream>>>(
      zout, Wn, nullptr, xhat, B_, D_, P_, P_, P_, D_);
}